// CrossGAT_60232621359630
// MI455X (gfx1250) — compile-verified
//
#include <hip/hip_runtime.h>
#include <math.h>

#define N_  16000
#define E_  256000
#define D_  128
#define H_  4
#define HD_ (H_ * D_)      // 512
#define G_  (3 * D_)       // 384
#define ALPHA_ 0.2f

typedef __attribute__((ext_vector_type(2))) float v2f;
typedef __attribute__((ext_vector_type(8))) float v8f;

// -------------------------------------------------------------------------
// K1: Wh[n,h,:] = h[n,:] @ W[h]   via V_WMMA_F32_16X16X4_F32
// grid (N/16, H), block 32 (one wave). 8 col-tiles x 32 k-steps = 256 wmma.
// -------------------------------------------------------------------------
__global__ void wh_kernel(const float* __restrict__ hfeat,
                          const float* __restrict__ W,
                          float* __restrict__ Wh) {
    const int n0   = blockIdx.x * 16;
    const int head = blockIdx.y;
    const int lane = threadIdx.x;          // 0..31
    const int row  = lane & 15;
    const int khalf = (lane >> 4) * 2;     // lanes 0-15 -> K {k,k+1}; 16-31 -> {k+2,k+3}
    const float* __restrict__ Wp = W + head * D_ * D_;

    v8f c[8];
    #pragma unroll
    for (int t = 0; t < 8; ++t)
        #pragma unroll
        for (int r = 0; r < 8; ++r) c[t][r] = 0.0f;

    for (int k = 0; k < D_; k += 4) {
        v2f a;
        const float* hp = hfeat + (n0 + row) * D_ + k + khalf;
        a.x = hp[0];
        a.y = hp[1];
        #pragma unroll
        for (int t = 0; t < 8; ++t) {
            v2f b;
            const float* bp = Wp + (k + khalf) * D_ + t * 16 + row;
            b.x = bp[0];     // B[k+khalf][col]
            b.y = bp[D_];    // B[k+khalf+1][col]
            c[t] = __builtin_amdgcn_wmma_f32_16x16x4_f32(
                false, a, false, b, (short)0, c[t], false, false);
        }
    }

    const int mbase = (lane < 16) ? 0 : 8;   // C vgpr r holds rows r / r+8
    #pragma unroll
    for (int t = 0; t < 8; ++t)
        #pragma unroll
        for (int r = 0; r < 8; ++r)
            Wh[(size_t)(n0 + mbase + r) * HD_ + head * D_ + t * 16 + row] = c[t][r];
}

// -------------------------------------------------------------------------
// K2: s_src[n,h] = Wh[n,h,:].a1[h],  s_dst[n,h] = Wh[n,h,:].a2[h]
// one wave per (n,h); each lane covers 4 of the 128 elements.
// -------------------------------------------------------------------------
__global__ void score_kernel(const float* __restrict__ Wh,
                             const float* __restrict__ a,
                             float* __restrict__ s_src,
                             float* __restrict__ s_dst) {
    const int gw = blockIdx.x * (blockDim.x >> 5) + (threadIdx.x >> 5);
    if (gw >= N_ * H_) return;
    const int n = gw / H_, head = gw % H_;
    const int lane = threadIdx.x & 31;
    const float* wp = Wh + (size_t)n * HD_ + head * D_;
    const float* a1 = a + head * 2 * D_;
    const float* a2 = a1 + D_;
    float ss = 0.f, sd = 0.f;
    #pragma unroll
    for (int i = 0; i < 4; ++i) {
        const int d = lane * 4 + i;
        const float w = wp[d];
        ss += w * a1[d];
        sd += w * a2[d];
    }
    #pragma unroll
    for (int off = 16; off > 0; off >>= 1) {
        ss += __shfl_down(ss, off, 32);
        sd += __shfl_down(sd, off, 32);
    }
    if (lane == 0) { s_src[gw] = ss; s_dst[gw] = sd; }
}

// -------------------------------------------------------------------------
// K3: init m=-inf, denom=0, h_prime=0 (workspace is poisoned by harness)
// -------------------------------------------------------------------------
__global__ void init_kernel(float* __restrict__ m,
                            float* __restrict__ denom,
                            float* __restrict__ h_prime) {
    const size_t i = (size_t)blockIdx.x * blockDim.x + threadIdx.x;
    if (i < (size_t)N_ * H_) { m[i] = -__builtin_inff(); denom[i] = 0.0f; }
    if (i < (size_t)N_ * HD_) h_prime[i] = 0.0f;
}

// -------------------------------------------------------------------------
// K4: per-(edge,head) leaky-relu score, segment max via float atomic max
// -------------------------------------------------------------------------
__global__ void edge_max_kernel(const float* __restrict__ s_src,
                                const float* __restrict__ s_dst,
                                const int* __restrict__ src,
                                const int* __restrict__ dst,
                                float* __restrict__ m) {
    const int i = blockIdx.x * blockDim.x + threadIdx.x;   // e*H + head
    if (i >= E_ * H_) return;
    const int e = i / H_, head = i % H_;
    const int s = src[e], t = dst[e];
    float v = s_src[s * H_ + head] + s_dst[t * H_ + head];
    v = (v > 0.0f) ? v : ALPHA_ * v;
    __hip_atomic_fetch_max(&m[t * H_ + head], v,
                           __ATOMIC_RELAXED, __HIP_MEMORY_SCOPE_AGENT);
}

// -------------------------------------------------------------------------
// K5: ex = exp(e - m[dst]); denom[dst] += ex  (atomic f32 add)
// -------------------------------------------------------------------------
__global__ void edge_exp_kernel(const float* __restrict__ s_src,
                                const float* __restrict__ s_dst,
                                const int* __restrict__ src,
                                const int* __restrict__ dst,
                                const float* __restrict__ m,
                                float* __restrict__ ex,
                                float* __restrict__ denom) {
    const int i = blockIdx.x * blockDim.x + threadIdx.x;   // e*H + head
    if (i >= E_ * H_) return;
    const int e = i / H_, head = i % H_;
    const int s = src[e], t = dst[e];
    float v = s_src[s * H_ + head] + s_dst[t * H_ + head];
    v = (v > 0.0f) ? v : ALPHA_ * v;
    const float exv = expf(v - m[t * H_ + head]);
    ex[i] = exv;
    __hip_atomic_fetch_add(&denom[t * H_ + head], exv,
                           __ATOMIC_RELAXED, __HIP_MEMORY_SCOPE_AGENT);
}

// -------------------------------------------------------------------------
// K6: h_prime[dst] += att * Wh[src]   (block per edge, 128 threads, 4 heads)
// Wh (32MB) and h_prime (32MB) are L2-resident on MI455X (192MB L2).
// -------------------------------------------------------------------------
__global__ void scatter_kernel(const float* __restrict__ Wh,
                               const float* __restrict__ ex,
                               const float* __restrict__ denom,
                               const int* __restrict__ src,
                               const int* __restrict__ dst,
                               float* __restrict__ h_prime) {
    const int e = blockIdx.x;
    const int d = threadIdx.x;           // 0..127
    const int s = src[e], t = dst[e];
    #pragma unroll
    for (int head = 0; head < H_; ++head) {
        const float att = ex[e * H_ + head] / denom[t * H_ + head];
        const float val = att * Wh[(size_t)(s * H_ + head) * D_ + d];
        __hip_atomic_fetch_add(&h_prime[(size_t)(t * H_ + head) * D_ + d], val,
                               __ATOMIC_RELAXED, __HIP_MEMORY_SCOPE_AGENT);
    }
}

// -------------------------------------------------------------------------
// K7: GRU.  gi = h' @ W_ih^T (K=512), gh = h @ W_hh^T (K=128) via f32 WMMA,
// then gates + blend.  grid (N/16, D/16), block 32.
// gate g of output col-tile j needs gi/gh columns {j, j+8, j+16} (x16).
// -------------------------------------------------------------------------
__global__ void gru_kernel(const float* __restrict__ hfeat,
                           const float* __restrict__ h_prime,
                           const float* __restrict__ W_ih,   // (384, 512)
                           const float* __restrict__ W_hh,   // (384, 128)
                           const float* __restrict__ b_ih,
                           const float* __restrict__ b_hh,
                           float* __restrict__ out) {
    const int n0   = blockIdx.x * 16;
    const int j    = blockIdx.y;           // output col tile, cols j*16..j*16+15
    const int lane = threadIdx.x;
    const int row  = lane & 15;
    const int khalf = (lane >> 4) * 2;

    v8f gi[3], gh[3];
    #pragma unroll
    for (int g = 0; g < 3; ++g)
        #pragma unroll
        for (int r = 0; r < 8; ++r) { gi[g][r] = 0.0f; gh[g][r] = 0.0f; }

    // gi accumulation over x = h_prime (N, 512)
    for (int k = 0; k < HD_; k += 4) {
        v2f a;
        const float* ap = h_prime + (size_t)(n0 + row) * HD_ + k + khalf;
        a.x = ap[0];
        a.y = ap[1];
        #pragma unroll
        for (int g = 0; g < 3; ++g) {
            const int c = g * D_ + j * 16 + row;       // gate column (0..383)
            v2f b;                                     // B[kk][cc] = W_ih[c][k+kk]
            const float* bp = W_ih + (size_t)c * HD_ + k + khalf;
            b.x = bp[0];
            b.y = bp[1];
            gi[g] = __builtin_amdgcn_wmma_f32_16x16x4_f32(
                false, a, false, b, (short)0, gi[g], false, false);
        }
    }

    // gh accumulation over h (N, 128)
    for (int k = 0; k < D_; k += 4) {
        v2f a;
        const float* ap = hfeat + (size_t)(n0 + row) * D_ + k + khalf;
        a.x = ap[0];
        a.y = ap[1];
        #pragma unroll
        for (int g = 0; g < 3; ++g) {
            const int c = g * D_ + j * 16 + row;
            v2f b;
            const float* bp = W_hh + (size_t)c * D_ + k + khalf;
            b.x = bp[0];
            b.y = bp[1];
            gh[g] = __builtin_amdgcn_wmma_f32_16x16x4_f32(
                false, a, false, b, (short)0, gh[g], false, false);
        }
    }

    // gates + blend
    const int mbase = (lane < 16) ? 0 : 8;
    const int col = j * 16 + row;
    #pragma unroll
    for (int r = 0; r < 8; ++r) {
        const int nrow = n0 + mbase + r;
        const float ir = gi[0][r] + b_ih[col];
        const float iz = gi[1][r] + b_ih[D_ + col];
        const float in_ = gi[2][r] + b_ih[2 * D_ + col];
        const float hr = gh[0][r] + b_hh[col];
        const float hz = gh[1][r] + b_hh[D_ + col];
        const float hn = gh[2][r] + b_hh[2 * D_ + col];
        const float rg = 1.0f / (1.0f + expf(-(ir + hr)));
        const float zg = 1.0f / (1.0f + expf(-(iz + hz)));
        const float nv = tanhf(in_ + rg * hn);
        const float hv = hfeat[(size_t)nrow * D_ + col];
        out[(size_t)nrow * D_ + col] = (1.0f - zg) * nv + zg * hv;
    }
}

// -------------------------------------------------------------------------
extern "C" void kernel_launch(void* const* d_in, const int* in_sizes, int n_in,
                              void* d_out, int out_size, void* d_ws, size_t ws_size,
                              hipStream_t stream) {
    const float* hfeat = (const float*)d_in[0];
    const float* W     = (const float*)d_in[1];
    const float* a     = (const float*)d_in[2];
    const float* W_ih  = (const float*)d_in[3];
    const float* W_hh  = (const float*)d_in[4];
    const float* b_ih  = (const float*)d_in[5];
    const float* b_hh  = (const float*)d_in[6];
    const int*   src   = (const int*)d_in[7];
    const int*   dst   = (const int*)d_in[8];
    float* out = (float*)d_out;

    // workspace carve-up (~70.6 MB)
    float* ws      = (float*)d_ws;
    float* Wh      = ws;                              // N*H*D
    float* s_src   = Wh + (size_t)N_ * HD_;           // N*H
    float* s_dst   = s_src + (size_t)N_ * H_;         // N*H
    float* m       = s_dst + (size_t)N_ * H_;         // N*H
    float* denom   = m + (size_t)N_ * H_;             // N*H
    float* ex      = denom + (size_t)N_ * H_;         // E*H
    float* h_prime = ex + (size_t)E_ * H_;            // N*H*D

    wh_kernel<<<dim3(N_ / 16, H_), 32, 0, stream>>>(hfeat, W, Wh);

    score_kernel<<<(N_ * H_ + 7) / 8, 256, 0, stream>>>(Wh, a, s_src, s_dst);

    init_kernel<<<((size_t)N_ * HD_ + 255) / 256, 256, 0, stream>>>(m, denom, h_prime);

    edge_max_kernel<<<(E_ * H_ + 255) / 256, 256, 0, stream>>>(s_src, s_dst, src, dst, m);

    edge_exp_kernel<<<(E_ * H_ + 255) / 256, 256, 0, stream>>>(s_src, s_dst, src, dst,
                                                               m, ex, denom);

    scatter_kernel<<<E_, D_, 0, stream>>>(Wh, ex, denom, src, dst, h_prime);

    gru_kernel<<<dim3(N_ / 16, D_ / 16), 32, 0, stream>>>(hfeat, h_prime, W_ih, W_hh,
                                                          b_ih, b_hh, out);
}